// Weighted_adj_19413252178579
// MI455X (gfx1250) — compile-verified
//
#include <hip/hip_runtime.h>

// Problem constants (from the reference)
constexpr int kB = 8;
constexpr int kN = 512;
constexpr int kC = 128;
constexpr int kSel = (kN * kN) / 6;              // 43690 largest values -> 1.0
constexpr float kScale = 0.08838834764831845f;   // (128)^-0.5

typedef float v2f __attribute__((ext_vector_type(2)));
typedef float v8f __attribute__((ext_vector_type(8)));

// CDNA5 fp32 matrix op: D(16x16) = A(16x4) * B(4x16) + C, one matrix per wave32.
__device__ __forceinline__ v8f wmma4(v2f a, v2f b, v8f c) {
  return __builtin_amdgcn_wmma_f32_16x16x4_f32(
      /*neg_a=*/false, a, /*neg_b=*/false, b,
      /*c_mod=*/(short)0, c, /*reuse_a=*/false, /*reuse_b=*/false);
}

// Monotone float->uint mapping (total order matches float compare)
__device__ __forceinline__ unsigned f32_sortable(float f) {
  unsigned u = __float_as_uint(f);
  return (u & 0x80000000u) ? ~u : (u | 0x80000000u);
}

// -------- per-row stats: sq = sum x^2, mean, dinv = 1/sqrt(sum (x-m)^2) -----
__global__ void k_stats(const float* __restrict__ x, float* __restrict__ sq,
                        float* __restrict__ mean, float* __restrict__ dinv) {
  const int row  = blockIdx.x * 8 + (threadIdx.x >> 5);
  const int lane = threadIdx.x & 31;
  const float4 v = ((const float4*)(x + (size_t)row * kC))[lane];
  float s  = v.x + v.y + v.z + v.w;
  float ss = v.x * v.x + v.y * v.y + v.z * v.z + v.w * v.w;
  for (int off = 16; off >= 1; off >>= 1) {
    s  += __shfl_xor(s, off, 32);
    ss += __shfl_xor(ss, off, 32);
  }
  if (lane == 0) {
    float m   = s * (1.0f / kC);
    float var = ss - (float)kC * m * m;
    sq[row]   = ss;
    mean[row] = m;
    dinv[row] = 1.0f / sqrtf(fmaxf(var, 1e-30f));
  }
}

// -------- E/Ch/Corr tiles: WMMA gram + fused chebyshev --------------------
// grid (N/16, N/16, B), block = 32 (one wave per 16x16 tile)
__global__ void k_adj(const float* __restrict__ x,
                      const float* __restrict__ sq, const float* __restrict__ mean,
                      const float* __restrict__ dinv,
                      float* __restrict__ E, float* __restrict__ Ch,
                      float* __restrict__ Cr) {
  __shared__ float rowT[16][132];   // stride 132 -> 4 mod 64 banks, conflict-free
  __shared__ float colT[16][132];
  const int b    = blockIdx.z;
  const int R    = blockIdx.y * 16;
  const int Cc   = blockIdx.x * 16;
  const int lane = threadIdx.x;
  const int half = lane >> 4;
  const int lr   = lane & 15;
  const float* xb = x + (size_t)b * kN * kC;

  // stage 16x128 row tile + 16x128 col tile (float4 coalesced)
  for (int i = lane; i < 512; i += 32) {
    int r = i >> 5, q = i & 31;
    float4 a = ((const float4*)(xb + (size_t)(R + r) * kC))[q];
    rowT[r][q * 4 + 0] = a.x; rowT[r][q * 4 + 1] = a.y;
    rowT[r][q * 4 + 2] = a.z; rowT[r][q * 4 + 3] = a.w;
    float4 c = ((const float4*)(xb + (size_t)(Cc + r) * kC))[q];
    colT[r][q * 4 + 0] = c.x; colT[r][q * 4 + 1] = c.y;
    colT[r][q * 4 + 2] = c.z; colT[r][q * 4 + 3] = c.w;
  }
  __syncthreads();

  // gram tile via f32 WMMA, K = 128 in steps of 4.
  // Cap unroll at 8: full unroll (32 steps) made the compiler pipeline every
  // LDS operand fetch and spill VGPRs to scratch.
  v8f acc = {};
#pragma unroll 8
  for (int kb = 0; kb < kC; kb += 4) {
    v2f a, bb;
    a.x  = rowT[lr][kb + 2 * half]; a.y  = rowT[lr][kb + 2 * half + 1];
    bb.x = colT[lr][kb + 2 * half]; bb.y = colT[lr][kb + 2 * half + 1];
    acc = wmma4(a, bb, acc);
  }

  // fused chebyshev: max_c |xrow[m][c] - xcol[n][c]| for this lane's 8 outputs
  float cheb[8] = {0.f, 0.f, 0.f, 0.f, 0.f, 0.f, 0.f, 0.f};
#pragma unroll 4
  for (int c = 0; c < kC; ++c) {
    float xc = colT[lr][c];
#pragma unroll
    for (int v = 0; v < 8; ++v)
      cheb[v] = fmaxf(cheb[v], fabsf(rowT[v + 8 * half][c] - xc));
  }

  // epilogue: euclidean + corrcoef (cov = gram - C*mi*mj)
  const int base = b * kN;
  const float sqc = sq[base + Cc + lr];
  const float mc  = mean[base + Cc + lr];
  const float dc  = dinv[base + Cc + lr];
#pragma unroll
  for (int v = 0; v < 8; ++v) {
    int m = v + 8 * half;
    float g  = acc[v];
    float d2 = sq[base + R + m] + sqc - 2.0f * g;
    float e  = sqrtf(fmaxf(d2, 0.0f));
    float cv = (g - (float)kC * mean[base + R + m] * mc) * dinv[base + R + m] * dc;
    cv = fminf(1.0f, fmaxf(-1.0f, cv));
    size_t o = ((size_t)(base + R + m)) * kN + Cc + lr;
    E[o]  = e;
    Cr[o] = cv;
    Ch[o] = cheb[v];
  }
}

// -------- logits = (E * Ch^T) * scale : A[MxK] * B[NxK]^T ------------------
__global__ void k_logits(const float* __restrict__ A, const float* __restrict__ Bm,
                         float* __restrict__ Out) {
  __shared__ float At[16][68];
  __shared__ float Bt[16][68];
  const int b = blockIdx.z, R = blockIdx.y * 16, Cc = blockIdx.x * 16;
  const int lane = threadIdx.x, half = lane >> 4, lr = lane & 15;
  const float* Ab = A  + (size_t)b * kN * kN;
  const float* Bb = Bm + (size_t)b * kN * kN;
  v8f acc = {};
  for (int k0 = 0; k0 < kN; k0 += 64) {
    for (int i = lane; i < 256; i += 32) {
      int r = i >> 4, q = i & 15;
      float4 a = ((const float4*)(Ab + (size_t)(R + r) * kN + k0))[q];
      At[r][q * 4 + 0] = a.x; At[r][q * 4 + 1] = a.y;
      At[r][q * 4 + 2] = a.z; At[r][q * 4 + 3] = a.w;
      float4 c = ((const float4*)(Bb + (size_t)(Cc + r) * kN + k0))[q];
      Bt[r][q * 4 + 0] = c.x; Bt[r][q * 4 + 1] = c.y;
      Bt[r][q * 4 + 2] = c.z; Bt[r][q * 4 + 3] = c.w;
    }
    __syncthreads();
#pragma unroll 8
    for (int kk = 0; kk < 64; kk += 4) {
      v2f a, bb;
      a.x  = At[lr][kk + 2 * half]; a.y  = At[lr][kk + 2 * half + 1];
      bb.x = Bt[lr][kk + 2 * half]; bb.y = Bt[lr][kk + 2 * half + 1];
      acc = wmma4(a, bb, acc);
    }
    __syncthreads();
  }
#pragma unroll
  for (int v = 0; v < 8; ++v) {
    size_t o = ((size_t)(b * kN + R + v + 8 * half)) * kN + Cc + lr;
    Out[o] = acc[v] * kScale;
  }
}

// -------- row softmax in place (one wave per 512-wide row) -----------------
__global__ void k_softmax(float* __restrict__ L) {
  const int row  = blockIdx.x * 8 + (threadIdx.x >> 5);
  const int lane = threadIdx.x & 31;
  float* p = L + (size_t)row * kN;
  float v[16];
#pragma unroll
  for (int i = 0; i < 16; ++i) v[i] = p[lane + i * 32];
  float mx = -3.402823466e38f;
#pragma unroll
  for (int i = 0; i < 16; ++i) mx = fmaxf(mx, v[i]);
  for (int off = 16; off >= 1; off >>= 1) mx = fmaxf(mx, __shfl_xor(mx, off, 32));
  float s = 0.0f;
#pragma unroll
  for (int i = 0; i < 16; ++i) { v[i] = expf(v[i] - mx); s += v[i]; }
  for (int off = 16; off >= 1; off >>= 1) s += __shfl_xor(s, off, 32);
  const float inv = 1.0f / s;
#pragma unroll
  for (int i = 0; i < 16; ++i) p[lane + i * 32] = v[i] * inv;
}

// -------- weighted = attn * Corr : A[MxK] * B[KxN] -------------------------
__global__ void k_weighted(const float* __restrict__ A, const float* __restrict__ Bm,
                           float* __restrict__ Out) {
  __shared__ float At[16][68];
  const int b = blockIdx.z, R = blockIdx.y * 16, Cc = blockIdx.x * 16;
  const int lane = threadIdx.x, half = lane >> 4, lr = lane & 15;
  const float* Ab = A  + (size_t)b * kN * kN;
  const float* Bb = Bm + (size_t)b * kN * kN;
  v8f acc = {};
  for (int k0 = 0; k0 < kN; k0 += 64) {
    for (int i = lane; i < 256; i += 32) {
      int r = i >> 4, q = i & 15;
      float4 a = ((const float4*)(Ab + (size_t)(R + r) * kN + k0))[q];
      At[r][q * 4 + 0] = a.x; At[r][q * 4 + 1] = a.y;
      At[r][q * 4 + 2] = a.z; At[r][q * 4 + 3] = a.w;
    }
    __syncthreads();
#pragma unroll 8
    for (int kk = 0; kk < 64; kk += 4) {
      const int kg = k0 + kk + 2 * half;     // B rows striped across halves
      v2f a, bb;
      a.x  = At[lr][kk + 2 * half]; a.y = At[lr][kk + 2 * half + 1];
      bb.x = Bb[(size_t)kg * kN + Cc + lr];
      bb.y = Bb[(size_t)(kg + 1) * kN + Cc + lr];
      acc = wmma4(a, bb, acc);
    }
    __syncthreads();
  }
#pragma unroll
  for (int v = 0; v < 8; ++v) {
    size_t o = ((size_t)(b * kN + R + v + 8 * half)) * kN + Cc + lr;
    Out[o] = acc[v];
  }
}

// -------- top-k select: two-level 16-bit radix histogram -------------------
__global__ void k_hist_hi(const float* __restrict__ W, unsigned* __restrict__ hist) {
  size_t i = (size_t)blockIdx.x * blockDim.x + threadIdx.x;  // exact B*N*N threads
  unsigned u = f32_sortable(W[i]);
  atomicAdd(&hist[((i >> 18) << 16) | (u >> 16)], 1u);       // N*N == 2^18
}

__global__ void k_scan_hi(const unsigned* __restrict__ hist,
                          unsigned* __restrict__ selBin, unsigned* __restrict__ above) {
  __shared__ unsigned csum[256];
  __shared__ unsigned suf[256];
  const int b = blockIdx.x, tid = threadIdx.x;
  const unsigned* h = hist + ((size_t)b << 16);
  unsigned tot = 0;
  for (int q = 0; q < 256; ++q) tot += h[tid * 256 + q];
  csum[tid] = tot;
  __syncthreads();
  if (tid == 0) {
    unsigned run = 0;
    for (int j = 255; j >= 0; --j) { suf[j] = run; run += csum[j]; }
  }
  __syncthreads();
  unsigned running = suf[tid];                    // count in bins above this chunk
  for (int q = 255; q >= 0; --q) {
    unsigned c = h[tid * 256 + q];
    if (running < (unsigned)kSel && running + c >= (unsigned)kSel) {
      selBin[b] = (unsigned)(tid * 256 + q);
      above[b]  = running;
    }
    running += c;
  }
}

__global__ void k_hist_lo(const float* __restrict__ W, const unsigned* __restrict__ selBin,
                          unsigned* __restrict__ hist2) {
  size_t i = (size_t)blockIdx.x * blockDim.x + threadIdx.x;
  const int b = (int)(i >> 18);
  unsigned u = f32_sortable(W[i]);
  if ((u >> 16) == selBin[b])
    atomicAdd(&hist2[((size_t)b << 16) | (u & 0xFFFFu)], 1u);
}

__global__ void k_scan_lo(const unsigned* __restrict__ hist2, const unsigned* __restrict__ selBin,
                          const unsigned* __restrict__ above, unsigned* __restrict__ Tval,
                          unsigned* __restrict__ rInc) {
  __shared__ unsigned csum[256];
  __shared__ unsigned suf[256];
  const int b = blockIdx.x, tid = threadIdx.x;
  const unsigned* h = hist2 + ((size_t)b << 16);
  const unsigned Kp = (unsigned)kSel - above[b];  // still needed inside the bin
  unsigned tot = 0;
  for (int q = 0; q < 256; ++q) tot += h[tid * 256 + q];
  csum[tid] = tot;
  __syncthreads();
  if (tid == 0) {
    unsigned run = 0;
    for (int j = 255; j >= 0; --j) { suf[j] = run; run += csum[j]; }
  }
  __syncthreads();
  unsigned running = suf[tid];
  for (int q = 255; q >= 0; --q) {
    unsigned c = h[tid * 256 + q];
    if (running < Kp && running + c >= Kp) {
      Tval[b] = (selBin[b] << 16) | (unsigned)(tid * 256 + q);
      rInc[b] = Kp - running;                     // equal-valued entries to include
    }
    running += c;
  }
}

__global__ void k_mask(const float* __restrict__ W, const unsigned* __restrict__ Tval,
                       const unsigned* __restrict__ rInc, unsigned* __restrict__ eqCnt,
                       float* __restrict__ out) {
  size_t i = (size_t)blockIdx.x * blockDim.x + threadIdx.x;
  const int b = (int)(i >> 18);
  unsigned u = f32_sortable(W[i]);
  const unsigned T = Tval[b];
  float o = 0.0f;
  if (u > T) {
    o = 1.0f;
  } else if (u == T) {
    if (atomicAdd(&eqCnt[b], 1u) < rInc[b]) o = 1.0f;
  }
  out[i] = o;
}

// ---------------------------------------------------------------------------
extern "C" void kernel_launch(void* const* d_in, const int* in_sizes, int n_in,
                              void* d_out, int out_size, void* d_ws, size_t ws_size,
                              hipStream_t stream) {
  (void)in_sizes; (void)n_in; (void)out_size; (void)ws_size;
  const float* x = (const float*)d_in[0];
  float* out = (float*)d_out;
  float* f = (float*)d_ws;
  const size_t S = (size_t)kB * kN * kN;

  float* E  = f;            // 8 MB  (later reused as weighted W)
  float* Ch = f + S;        // 8 MB
  float* Cr = f + 2 * S;    // 8 MB
  float* L  = f + 3 * S;    // 8 MB  (logits -> attn in place)
  float* W  = E;            // E dead after k_logits -> reuse
  float* sq   = f + 4 * S;
  float* mean = sq + (size_t)kB * kN;
  float* dinv = mean + (size_t)kB * kN;
  unsigned* hist   = (unsigned*)(dinv + (size_t)kB * kN);
  unsigned* hist2  = hist + ((size_t)kB << 16);
  unsigned* selBin = hist2 + ((size_t)kB << 16);
  unsigned* above  = selBin + kB;
  unsigned* Tval   = above + kB;
  unsigned* rInc   = Tval + kB;
  unsigned* eqCnt  = rInc + kB;

  const size_t clearBytes =
      (size_t)((eqCnt + kB) - hist) * sizeof(unsigned);   // hists + sel arrays
  hipMemsetAsync(hist, 0, clearBytes, stream);

  const dim3 tg(kN / 16, kN / 16, kB);
  k_stats   <<<kB * kN / 8, 256, 0, stream>>>(x, sq, mean, dinv);
  k_adj     <<<tg, 32, 0, stream>>>(x, sq, mean, dinv, E, Ch, Cr);
  k_logits  <<<tg, 32, 0, stream>>>(E, Ch, L);
  k_softmax <<<kB * kN / 8, 256, 0, stream>>>(L);
  k_weighted<<<tg, 32, 0, stream>>>(L, Cr, W);

  const int nblk = (int)(S / 256);
  k_hist_hi <<<nblk, 256, 0, stream>>>(W, hist);
  k_scan_hi <<<kB, 256, 0, stream>>>(hist, selBin, above);
  k_hist_lo <<<nblk, 256, 0, stream>>>(W, selBin, hist2);
  k_scan_lo <<<kB, 256, 0, stream>>>(hist2, selBin, above, Tval, rInc);
  k_mask    <<<nblk, 256, 0, stream>>>(W, Tval, rInc, eqCnt, out);
}